// AEVComputerOnyx_7189775253987
// MI455X (gfx1250) — compile-verified
//
#include <hip/hip_runtime.h>

// ---------------------------------------------------------------------------
// ANI AEV computer for gfx1250 (MI455X).
//  - Radial AEV: one-hot-over-species contraction done with
//    V_WMMA_F32_16X16X4_F32 accumulation (12 chunks of 4 neighbor atoms).
//  - Angular AEV: RCA neighbor-list compaction (ballot+popc, wave32), then
//    per-pair closed-form cos(theta - shfz) = x*cz + sqrt(1-x^2)*sz and
//    ds_add_f32 LDS atomics into the 10 species-pair channels.
//  - All transcendentals use raw gfx1250 hardware ops (v_cos_f32 via
//    __builtin_amdgcn_cosf takes revolutions: cos(2*pi*u); v_sqrt_f32;
//    v_exp_f32 via __expf) -- arguments are range-bounded here, so the
//    libm Payne-Hanek slow paths are avoided entirely.
// Block = 256 threads = 8 waves; each wave handles one center atom.
// Grid = (N molecules, 48/8).
// ---------------------------------------------------------------------------

#define A_ATOMS 48
#define NSPEC   4
#define NPAIRS  10
#define RAD_F   16            // SHFR count
#define ANG_F   32            // 4 SHFA * 8 SHFZ
#define AEV_LEN (NSPEC * RAD_F + NPAIRS * ANG_F)   // 64 + 320 = 384

typedef __attribute__((ext_vector_type(2))) float v2f;
typedef __attribute__((ext_vector_type(8))) float v8f;

__device__ __forceinline__ int triu_idx(int a, int b) {
  int s1 = a < b ? a : b;
  int s2 = a < b ? b : a;
  return s1 * NSPEC - (s1 * (s1 - 1)) / 2 + (s2 - s1);
}

// cos/sin of SHFZ[z] = (2z+1)*pi/16, compile-time constants.
__constant__ float CZ[8] = { 0.9807852804f,  0.8314696123f,  0.5555702330f,  0.1950903220f,
                            -0.1950903220f, -0.5555702330f, -0.8314696123f, -0.9807852804f };
__constant__ float SZ[8] = { 0.1950903220f,  0.5555702330f,  0.8314696123f,  0.9807852804f,
                             0.9807852804f,  0.8314696123f,  0.5555702330f,  0.1950903220f };
__constant__ float SHFA_C[4] = { 0.9f, 1.55f, 2.2f, 2.85f };

// hardware cos: __builtin_amdgcn_cosf(u) = cos(2*pi*u)
__device__ __forceinline__ float cutoff_cos(float d, float inv_2rc) {
  return 0.5f * __builtin_amdgcn_cosf(d * inv_2rc) + 0.5f;
}

// One (j, f) radial entry: B value for WMMA, plus one-hot A value.
__device__ __forceinline__ void rad_ab(const float* s_c, const int* s_s,
                                       float cix, float ciy, float ciz,
                                       int i, bool validI, int j, int f, float shfr,
                                       float& aval, float& bval) {
  constexpr float RCR = 5.2f;
  constexpr float ETA_R = 16.0f;
  float dx = cix - s_c[j * 3 + 0];
  float dy = ciy - s_c[j * 3 + 1];
  float dz = ciz - s_c[j * 3 + 2];
  float d2 = dx * dx + dy * dy + dz * dz;
  float d  = __builtin_amdgcn_sqrtf(d2 > 0.0f ? d2 : 1.0f);   // v_sqrt_f32
  int   sj = s_s[j];
  bool  ok = validI && (sj >= 0) && (j != i) && (d <= RCR);
  float fc = cutoff_cos(d, 0.5f / RCR);
  float dd = d - shfr;
  float v  = 0.25f * __expf(-ETA_R * dd * dd) * fc;           // v_exp_f32 path
  bval = ok ? v : 0.0f;
  aval = (sj == f) ? 1.0f : 0.0f;   // one-hot row M=f; masked j killed by bval=0
}

__launch_bounds__(256)
__global__ void aev_kernel(const int* __restrict__ species,
                           const float* __restrict__ coords,
                           float* __restrict__ out) {
  constexpr float RCA = 3.5f;
  constexpr float ETA_A = 8.0f;

  const int n    = blockIdx.x;
  const int wave = threadIdx.x >> 5;
  const int lane = threadIdx.x & 31;
  const int i    = blockIdx.y * 8 + wave;   // center atom, always < 48

  __shared__ float s_c[A_ATOMS * 3];
  __shared__ int   s_s[A_ATOMS];
  __shared__ float s_nd [8][A_ATOMS];
  __shared__ float s_nfc[8][A_ATOMS];
  __shared__ float s_ndx[8][A_ATOMS];
  __shared__ float s_ndy[8][A_ATOMS];
  __shared__ float s_ndz[8][A_ATOMS];
  __shared__ int   s_ns [8][A_ATOMS];
  __shared__ float s_ang[8][NPAIRS * ANG_F];

  // ---- stage molecule into LDS ----
  for (int t = threadIdx.x; t < A_ATOMS * 3; t += blockDim.x)
    s_c[t] = coords[(size_t)n * (A_ATOMS * 3) + t];
  for (int t = threadIdx.x; t < A_ATOMS; t += blockDim.x)
    s_s[t] = species[n * A_ATOMS + t];
  __syncthreads();

  const float cix = s_c[i * 3 + 0];
  const float ciy = s_c[i * 3 + 1];
  const float ciz = s_c[i * 3 + 2];
  const bool  validI = (s_s[i] >= 0);

  // =====================  RADIAL via WMMA F32 16x16x4  =====================
  // D[s,f] = sum_j onehot(species_j == s) * rad[j,f]
  // A(16x4): lanes 0-15 hold M=lane, a.x -> K=0 (j0) / K=2 (j0+2 upper half),
  //          a.y -> K=1 / K=3.  B(4x16) mirrored: lane holds N=f for row K.
  const int   f    = lane & 15;
  const int   half = lane >> 4;
  const float shfr = 0.9f + 0.26875f * (float)f;
  v8f cacc = {};
  for (int j0 = 0; j0 < A_ATOMS; j0 += 4) {
    const int ja = j0 + half * 2;
    const int jb = ja + 1;
    float ax, ay, bx, by;
    rad_ab(s_c, s_s, cix, ciy, ciz, i, validI, ja, f, shfr, ax, bx);
    rad_ab(s_c, s_s, cix, ciy, ciz, i, validI, jb, f, shfr, ay, by);
    v2f a, b;
    a.x = ax; a.y = ay;
    b.x = bx; b.y = by;
    // 8 args: (neg_a, A, neg_b, B, c_mod, C, reuse_a, reuse_b)
    cacc = __builtin_amdgcn_wmma_f32_16x16x4_f32(false, a, false, b,
                                                 (short)0, cacc, false, false);
  }

  // =====================  ANGULAR: neighbor list  =====================
  for (int t = lane; t < NPAIRS * ANG_F; t += 32) s_ang[wave][t] = 0.0f;

  int cnt = 0;
  for (int base = 0; base < A_ATOMS; base += 32) {
    int  j  = base + lane;
    bool ok = false;
    float d = 0.f, dx = 0.f, dy = 0.f, dz = 0.f;
    int sj = -1;
    if (j < A_ATOMS && j != i) {
      sj = s_s[j];
      dx = cix - s_c[j * 3 + 0];
      dy = ciy - s_c[j * 3 + 1];
      dz = ciz - s_c[j * 3 + 2];
      float d2 = dx * dx + dy * dy + dz * dz;
      d = __builtin_amdgcn_sqrtf(d2 > 0.0f ? d2 : 1.0f);
      ok = validI && (sj >= 0) && (d <= RCA);
    }
    unsigned mask = (unsigned)__ballot(ok);
    int offs = __popc(mask & ((1u << lane) - 1u));
    if (ok) {
      int idx = cnt + offs;
      s_nd [wave][idx] = d;
      s_nfc[wave][idx] = cutoff_cos(d, 0.5f / RCA);
      s_ndx[wave][idx] = dx;
      s_ndy[wave][idx] = dy;
      s_ndz[wave][idx] = dz;
      s_ns [wave][idx] = sj;
    }
    cnt += __popc(mask);
  }
  __threadfence_block();   // wave-internal LDS ordering (compiler fence + dscnt)

  // =====================  ANGULAR: pair loop  =====================
  const int M = cnt;
  const int P = (M * (M - 1)) / 2;
  for (int t = lane; t < P; t += 32) {
    // decode upper-triangular pair index t -> (jj < kk)
    int jj = 0, rem = t;
    while (rem >= M - 1 - jj) { rem -= (M - 1 - jj); ++jj; }
    int kk = jj + 1 + rem;

    float dj = s_nd[wave][jj], dk = s_nd[wave][kk];
    float fcp2 = 2.0f * s_nfc[wave][jj] * s_nfc[wave][kk];
    float dot = s_ndx[wave][jj] * s_ndx[wave][kk]
              + s_ndy[wave][jj] * s_ndy[wave][kk]
              + s_ndz[wave][jj] * s_ndz[wave][kk];
    float denom = fmaxf(dj * dk, 1e-10f);
    float x  = 0.95f * dot / denom;                              // cos(theta)*0.95
    float st = __builtin_amdgcn_sqrtf(fmaxf(1.0f - x * x, 0.0f)); // sin(theta)
    float dh = 0.5f * (dj + dk);

    float f2[4];
#pragma unroll
    for (int a = 0; a < 4; ++a) {
      float dd = dh - SHFA_C[a];
      f2[a] = __expf(-ETA_A * dd * dd) * fcp2;
    }

    int p = triu_idx(s_ns[wave][jj], s_ns[wave][kk]);
    float* dst = &s_ang[wave][p * ANG_F];
#pragma unroll
    for (int z = 0; z < 8; ++z) {
      // ((1 + cos(theta - shfz))/2)^32 via 5 squarings
      float h  = 0.5f * (1.0f + x * CZ[z] + st * SZ[z]);
      float h2  = h * h;
      float h4  = h2 * h2;
      float h8  = h4 * h4;
      float h16 = h8 * h8;
      float h32 = h16 * h16;
#pragma unroll
      for (int a = 0; a < 4; ++a)
        atomicAdd(&dst[a * 8 + z], f2[a] * h32);    // ds_add_f32
    }
  }
  __threadfence_block();

  // =====================  write out  =====================
  float* outp = out + (size_t)(n * A_ATOMS + i) * AEV_LEN;
  if (lane < 16) {
    // C/D layout: VGPR r, lanes 0-15 -> (M=r, N=lane); rows 0..3 = species
    outp[0 * 16 + f] = cacc[0];
    outp[1 * 16 + f] = cacc[1];
    outp[2 * 16 + f] = cacc[2];
    outp[3 * 16 + f] = cacc[3];
  }
  for (int t = lane; t < NPAIRS * ANG_F; t += 32)
    outp[NSPEC * RAD_F + t] = s_ang[wave][t];
}

extern "C" void kernel_launch(void* const* d_in, const int* in_sizes, int n_in,
                              void* d_out, int out_size, void* d_ws, size_t ws_size,
                              hipStream_t stream) {
  const int*   species = (const int*)d_in[0];
  const float* coords  = (const float*)d_in[1];
  float*       outp    = (float*)d_out;
  const int N = in_sizes[0] / A_ATOMS;      // 32 molecules
  dim3 grid(N, A_ATOMS / 8);                // 8 waves/block, 1 atom per wave
  aev_kernel<<<grid, 256, 0, stream>>>(species, coords, outp);
}